// StrangMagHeatNet_52536039964972
// MI455X (gfx1250) — compile-verified
//
#include <hip/hip_runtime.h>
#include <math.h>

typedef __attribute__((ext_vector_type(2))) float v2f;
typedef __attribute__((ext_vector_type(8))) float v8f;

#define Nn   50000
#define NNZn 800000
#define IND  128
#define PP   64
#define HH   64
#define CC   40

// ---------------------------------------------------------------------------
// fp32 WMMA 16x16x4 wrapper (exact fp32 GEMM path on CDNA5)
// ---------------------------------------------------------------------------
__device__ __forceinline__ v8f wmma_f32(v2f a, v2f b, v8f c) {
  return __builtin_amdgcn_wmma_f32_16x16x4_f32(false, a, false, b, (short)0, c,
                                               false, false);
}

// ---------------------------------------------------------------------------
// Utility kernels
// ---------------------------------------------------------------------------
__global__ void zero_f32_k(float* __restrict__ p, int n) {
  for (int i = blockIdx.x * blockDim.x + threadIdx.x; i < n;
       i += gridDim.x * blockDim.x)
    p[i] = 0.f;
}

__global__ void zero_i32_k(int* __restrict__ p, int n) {
  for (int i = blockIdx.x * blockDim.x + threadIdx.x; i < n;
       i += gridDim.x * blockDim.x)
    p[i] = 0;
}

__global__ void init_logits_k(float* __restrict__ logits,
                              const float* __restrict__ bc) {
  int i = blockIdx.x * blockDim.x + threadIdx.x;
  if (i < Nn * CC) logits[i] = bc[i % CC];
}

// ---------------------------------------------------------------------------
// CSR build: histogram -> block scan -> scatter (done once per launch)
// ---------------------------------------------------------------------------
__global__ void hist_k(const int* __restrict__ rows, int* __restrict__ cnt,
                       int nnz) {
  int e = blockIdx.x * blockDim.x + threadIdx.x;
  if (e < nnz) atomicAdd(&cnt[rows[e]], 1);
}

// Exclusive scan of cnt[0..n-1]; writes rp[0..n] and overwrites cnt with the
// running (exclusive) start offsets, used as scatter cursors afterwards.
__global__ void scan_k(int* __restrict__ cnt, int* __restrict__ rp, int n) {
  __shared__ int sh[1024];
  __shared__ int carry;
  int tid = threadIdx.x;
  if (tid == 0) { carry = 0; rp[0] = 0; }
  __syncthreads();
  for (int base = 0; base < n; base += 1024) {
    int i = base + tid;
    int v = (i < n) ? cnt[i] : 0;
    sh[tid] = v;
    __syncthreads();
    for (int off = 1; off < 1024; off <<= 1) {
      int t = (tid >= off) ? sh[tid - off] : 0;
      __syncthreads();
      sh[tid] += t;
      __syncthreads();
    }
    int incl = sh[tid];
    int c = carry;
    if (i < n) {
      rp[i + 1] = c + incl;
      cnt[i] = c + incl - v;  // exclusive start -> scatter cursor
    }
    __syncthreads();
    if (tid == 0) carry = c + sh[1023];
    __syncthreads();
  }
}

__global__ void scatter_k(const int* __restrict__ rows,
                          const int* __restrict__ cols,
                          const float* __restrict__ vals,
                          int* __restrict__ cursor, int2* __restrict__ pairs,
                          int nnz) {
  int e = blockIdx.x * blockDim.x + threadIdx.x;
  if (e < nnz) {
    int r = rows[e];
    int p = atomicAdd(&cursor[r], 1);
    pairs[p] = make_int2(cols[e], __float_as_int(vals[e]));
  }
}

// ---------------------------------------------------------------------------
// Fused CSR SpMM + series update:  Tout = L @ Tin ; Y += coef * Tout
// One wave (32 lanes, float2 per lane = 64 cols) owns one output row.
// No atomics; gathers are L2-resident (12.8 MB working set vs 192 MB L2).
// ---------------------------------------------------------------------------
__global__ void spmm_axpy_k(const int* __restrict__ rp,
                            const int2* __restrict__ pairs,
                            const float* __restrict__ Tin,
                            float* __restrict__ Tout, float* __restrict__ Y,
                            float coef) {
  int row = blockIdx.x * blockDim.y + threadIdx.y;
  if (row >= Nn) return;
  int lane = threadIdx.x;
  int s = rp[row], e = rp[row + 1];
  float ax = 0.f, ay = 0.f;
  for (int j = s; j < e; ++j) {
    int2 p = pairs[j];                      // one b64 broadcast load per nnz
    float v = __int_as_float(p.y);
    const float2 x = *(const float2*)(Tin + (p.x << 6) + (lane << 1));
    ax = fmaf(v, x.x, ax);
    ay = fmaf(v, x.y, ay);
  }
  int o = (row << 6) + (lane << 1);
  float2 t; t.x = ax; t.y = ay;
  *(float2*)(Tout + o) = t;
  float2 y = *(float2*)(Y + o);
  y.x = fmaf(coef, ax, y.x);
  y.y = fmaf(coef, ay, y.y);
  *(float2*)(Y + o) = y;
}

// ---------------------------------------------------------------------------
// GEMM 1: pre = X[50000,128] @ Wp[128,64]   (bias+LN+ReLU done separately)
// One wave per 16x16 tile, fp32 WMMA, K steps of 4.
// A frag: lanes 0-15 row M=lane, K={k,k+1}; lanes 16-31 K={k+2,k+3}.
// ---------------------------------------------------------------------------
__global__ void proj_gemm_k(const float* __restrict__ X,
                            const float* __restrict__ W,
                            float* __restrict__ pre) {
  int lane = threadIdx.x & 31;
  int gw = blockIdx.x * (blockDim.x >> 5) + (threadIdx.x >> 5);
  const int MT = Nn / 16, NT = PP / 16;
  if (gw >= MT * NT) return;
  int mt = gw / NT, nt = gw % NT;
  int l15 = lane & 15;
  int khalf = (lane >> 4) << 1;
  int rowA = mt * 16 + l15;
  int col = nt * 16 + l15;
  v8f acc;
#pragma unroll
  for (int i = 0; i < 8; i++) acc[i] = 0.f;
  for (int k0 = 0; k0 < IND; k0 += 4) {
    int k = k0 + khalf;
    v2f a, b;
    a.x = X[rowA * IND + k];
    a.y = X[rowA * IND + k + 1];
    b.x = W[k * PP + col];
    b.y = W[(k + 1) * PP + col];
    acc = wmma_f32(a, b, acc);
  }
  int rb = mt * 16 + ((lane >> 4) << 3);
#pragma unroll
  for (int i = 0; i < 8; i++) pre[(rb + i) * PP + col] = acc[i];
}

// ---------------------------------------------------------------------------
// LayerNorm(pre + bp) * gamma + beta, then ReLU -> Xp. Wave per row, wave32
// shuffle reductions over the 64-wide row (float2 per lane).
// ---------------------------------------------------------------------------
__global__ void ln_relu_k(const float* __restrict__ pre,
                          const float* __restrict__ bp,
                          const float* __restrict__ gamma,
                          const float* __restrict__ beta,
                          float* __restrict__ Xp) {
  int row = blockIdx.x * blockDim.y + threadIdx.y;
  if (row >= Nn) return;
  int lane = threadIdx.x;
  int o = (row << 6) + (lane << 1);
  float2 x = *(const float2*)(pre + o);
  x.x += bp[lane * 2];
  x.y += bp[lane * 2 + 1];
  float s = x.x + x.y;
#pragma unroll
  for (int m = 16; m >= 1; m >>= 1) s += __shfl_xor(s, m, 32);
  float mu = s * (1.f / 64.f);
  float dx = x.x - mu, dy = x.y - mu;
  float q = dx * dx + dy * dy;
#pragma unroll
  for (int m = 16; m >= 1; m >>= 1) q += __shfl_xor(q, m, 32);
  float r = rsqrtf(q * (1.f / 64.f) + 1e-5f);
  float2 outv;
  outv.x = fmaxf(dx * r * gamma[lane * 2] + beta[lane * 2], 0.f);
  outv.y = fmaxf(dy * r * gamma[lane * 2 + 1] + beta[lane * 2 + 1], 0.f);
  *(float2*)(Xp + o) = outv;
}

// ---------------------------------------------------------------------------
// GEMM 2: h = relu( [Zr | Zi | Xp-Zr | -Zi] @ Wt + bt ), feat built on the fly
// ---------------------------------------------------------------------------
__device__ __forceinline__ float feat_elem(const float* __restrict__ Zr,
                                           const float* __restrict__ Zi,
                                           const float* __restrict__ Xp,
                                           int row, int k) {
  int q = k >> 6;
  int o = (row << 6) + (k & 63);
  if (q == 0) return Zr[o];
  if (q == 1) return Zi[o];
  if (q == 2) return Xp[o] - Zr[o];
  return -Zi[o];  // Xp_i == 0 exactly
}

__global__ void feat_gemm_k(const float* __restrict__ Zr,
                            const float* __restrict__ Zi,
                            const float* __restrict__ Xp,
                            const float* __restrict__ Wt,
                            const float* __restrict__ bt,
                            float* __restrict__ Hout) {
  int lane = threadIdx.x & 31;
  int gw = blockIdx.x * (blockDim.x >> 5) + (threadIdx.x >> 5);
  const int MT = Nn / 16, NT = HH / 16;
  if (gw >= MT * NT) return;
  int mt = gw / NT, nt = gw % NT;
  int l15 = lane & 15;
  int khalf = (lane >> 4) << 1;
  int rowA = mt * 16 + l15;
  int col = nt * 16 + l15;
  v8f acc;
#pragma unroll
  for (int i = 0; i < 8; i++) acc[i] = 0.f;
  for (int k0 = 0; k0 < 4 * PP; k0 += 4) {
    int k = k0 + khalf;
    v2f a, b;
    a.x = feat_elem(Zr, Zi, Xp, rowA, k);
    a.y = feat_elem(Zr, Zi, Xp, rowA, k + 1);
    b.x = Wt[k * HH + col];
    b.y = Wt[(k + 1) * HH + col];
    acc = wmma_f32(a, b, acc);
  }
  int rb = mt * 16 + ((lane >> 4) << 3);
  float bias = bt[col];
#pragma unroll
  for (int i = 0; i < 8; i++)
    Hout[(rb + i) * HH + col] = fmaxf(acc[i] + bias, 0.f);
}

// ---------------------------------------------------------------------------
// GEMM 3: logits[N,40] += h[N,64] @ Wc_slice[64,40]  (accumulate into d_out)
// ---------------------------------------------------------------------------
__global__ void out_gemm_k(const float* __restrict__ Hbuf,
                           const float* __restrict__ Wc,
                           float* __restrict__ logits) {
  int lane = threadIdx.x & 31;
  int gw = blockIdx.x * (blockDim.x >> 5) + (threadIdx.x >> 5);
  const int MT = Nn / 16, NT = 3;  // 48 >= 40 cols
  if (gw >= MT * NT) return;
  int mt = gw / NT, nt = gw % NT;
  int l15 = lane & 15;
  int khalf = (lane >> 4) << 1;
  int rowA = mt * 16 + l15;
  int col = nt * 16 + l15;
  bool cok = col < CC;
  int rb = mt * 16 + ((lane >> 4) << 3);
  v8f acc;
#pragma unroll
  for (int i = 0; i < 8; i++)
    acc[i] = cok ? logits[(rb + i) * CC + col] : 0.f;
  for (int k0 = 0; k0 < HH; k0 += 4) {
    int k = k0 + khalf;
    v2f a, b;
    a.x = Hbuf[rowA * HH + k];
    a.y = Hbuf[rowA * HH + k + 1];
    b.x = cok ? Wc[k * CC + col] : 0.f;
    b.y = cok ? Wc[(k + 1) * CC + col] : 0.f;
    acc = wmma_f32(a, b, acc);
  }
  if (cok) {
#pragma unroll
    for (int i = 0; i < 8; i++) logits[(rb + i) * CC + col] = acc[i];
  }
}

// ---------------------------------------------------------------------------
// Host orchestration
// ---------------------------------------------------------------------------
extern "C" void kernel_launch(void* const* d_in, const int* in_sizes, int n_in,
                              void* d_out, int out_size, void* d_ws,
                              size_t ws_size, hipStream_t stream) {
  (void)in_sizes; (void)n_in; (void)out_size; (void)ws_size;

  const float* X    = (const float*)d_in[0];
  const int*   symR = (const int*)d_in[1];
  const int*   symC = (const int*)d_in[2];
  const float* symV = (const float*)d_in[3];
  const int*   skR  = (const int*)d_in[4];
  const int*   skC  = (const int*)d_in[5];
  const float* skV  = (const float*)d_in[6];
  const float* Wp   = (const float*)d_in[7];
  const float* bp   = (const float*)d_in[8];
  const float* gamma= (const float*)d_in[9];
  const float* beta = (const float*)d_in[10];
  const float* Wt   = (const float*)d_in[11];
  const float* bt   = (const float*)d_in[12];
  const float* Wc   = (const float*)d_in[13];
  const float* bc   = (const float*)d_in[14];
  float* logits = (float*)d_out;

  const size_t NF = (size_t)Nn * PP;  // 3.2M floats per dense buffer
  float* ws  = (float*)d_ws;
  float* Xp  = ws;
  float* Zr  = Xp + NF;
  float* Zi  = Zr + NF;
  float* A0  = Zi + NF;   // term ping / pre-LN / h_t scratch
  float* A1  = A0 + NF;   // term pong
  int*  symRP  = (int*)(A1 + NF);
  int*  symCNT = symRP + (Nn + 1);
  int*  skRP   = symCNT + Nn;
  int*  skCNT  = skRP + (Nn + 1);
  int2* symPairs = (int2*)(skCNT + Nn);   // byte offset divisible by 8
  int2* skPairs  = symPairs + NNZn;

  // ---- CSR build for both operators (once per launch) ----
  zero_i32_k<<<256, 256, 0, stream>>>(symCNT, Nn);
  zero_i32_k<<<256, 256, 0, stream>>>(skCNT, Nn);
  hist_k<<<(NNZn + 255) / 256, 256, 0, stream>>>(symR, symCNT, NNZn);
  hist_k<<<(NNZn + 255) / 256, 256, 0, stream>>>(skR, skCNT, NNZn);
  scan_k<<<1, 1024, 0, stream>>>(symCNT, symRP, Nn);
  scan_k<<<1, 1024, 0, stream>>>(skCNT, skRP, Nn);
  scatter_k<<<(NNZn + 255) / 256, 256, 0, stream>>>(symR, symC, symV, symCNT,
                                                    symPairs, NNZn);
  scatter_k<<<(NNZn + 255) / 256, 256, 0, stream>>>(skR, skC, skV, skCNT,
                                                    skPairs, NNZn);

  // ---- projection + LayerNorm + ReLU ----
  const int projWaves = (Nn / 16) * (PP / 16);
  proj_gemm_k<<<(projWaves + 3) / 4, 128, 0, stream>>>(X, Wp, A0);
  ln_relu_k<<<(Nn + 7) / 8, dim3(32, 8), 0, stream>>>(A0, bp, gamma, beta, Xp);

  // ---- logits = bc (accumulated into per scale) ----
  init_logits_k<<<(Nn * CC + 255) / 256, 256, 0, stream>>>(logits, bc);

  double FH[13], FR[7];
  FH[0] = 1.0; for (int k = 1; k <= 12; k++) FH[k] = FH[k - 1] * k;
  FR[0] = 1.0; for (int k = 1; k <= 6; k++)  FR[k] = FR[k - 1] * k;
  const double SC[3] = {0.25, 0.5, 1.0};

  const size_t NB = NF * sizeof(float);
  dim3 spb(32, 8);
  const int spg = (Nn + 7) / 8;

  for (int ti = 0; ti < 3; ++ti) {
    double t = SC[ti];

    // Heat(t/2) on real channel (imag channel is exactly zero -> skipped)
    hipMemcpyAsync(Zr, Xp, NB, hipMemcpyDeviceToDevice, stream);
    hipMemcpyAsync(A0, Xp, NB, hipMemcpyDeviceToDevice, stream);
    float *tin = A0, *tout = A1;
    for (int k = 1; k <= 12; k++) {
      float c = (float)(pow(-0.5 * t, (double)k) / FH[k]);
      spmm_axpy_k<<<spg, spb, 0, stream>>>(symRP, symPairs, tin, tout, Zr, c);
      float* tmp = tin; tin = tout; tout = tmp;
    }

    // Rot(t): Ti channel is exactly zero, so single-channel series;
    // each k updates exactly one of {Zr, Zi} with +/- t^k/k!.
    zero_f32_k<<<4096, 256, 0, stream>>>(Zi, (int)NF);
    hipMemcpyAsync(A0, Zr, NB, hipMemcpyDeviceToDevice, stream);
    tin = A0; tout = A1;
    for (int k = 1; k <= 6; k++) {
      double tk = pow(t, (double)k) / FR[k];
      int rem = k & 3;
      float* Y = (rem == 1 || rem == 3) ? Zi : Zr;
      float c = (float)((rem == 1 || rem == 2) ? -tk : tk);
      spmm_axpy_k<<<spg, spb, 0, stream>>>(skRP, skPairs, tin, tout, Y, c);
      float* tmp = tin; tin = tout; tout = tmp;
    }

    // Heat(t/2) on Zr
    hipMemcpyAsync(A0, Zr, NB, hipMemcpyDeviceToDevice, stream);
    tin = A0; tout = A1;
    for (int k = 1; k <= 12; k++) {
      float c = (float)(pow(-0.5 * t, (double)k) / FH[k]);
      spmm_axpy_k<<<spg, spb, 0, stream>>>(symRP, symPairs, tin, tout, Zr, c);
      float* tmp = tin; tin = tout; tout = tmp;
    }
    // Heat(t/2) on Zi
    hipMemcpyAsync(A0, Zi, NB, hipMemcpyDeviceToDevice, stream);
    tin = A0; tout = A1;
    for (int k = 1; k <= 12; k++) {
      float c = (float)(pow(-0.5 * t, (double)k) / FH[k]);
      spmm_axpy_k<<<spg, spb, 0, stream>>>(symRP, symPairs, tin, tout, Zi, c);
      float* tmp = tin; tin = tout; tout = tmp;
    }

    // h_t = relu(feat @ Wt[ti] + bt[ti]) -> A0 ; logits += h_t @ Wc[ti rows]
    feat_gemm_k<<<(projWaves + 3) / 4, 128, 0, stream>>>(
        Zr, Zi, Xp, Wt + (size_t)ti * 4 * PP * HH, bt + ti * HH, A0);
    const int outWaves = (Nn / 16) * 3;
    out_gemm_k<<<(outWaves + 3) / 4, 128, 0, stream>>>(
        A0, Wc + (size_t)ti * HH * CC, logits);
  }
}